// InformationTheoreticQuaternionPool_11605001633859
// MI455X (gfx1250) — compile-verified
//
#include <hip/hip_runtime.h>
#include <hip/hip_bf16.h>

typedef __attribute__((ext_vector_type(2))) float v2f;
typedef __attribute__((ext_vector_type(8))) float v8f;

#define QP_EPS 1e-8f
#define LN2F   0.69314718055994530942f

// x layout: [b][g][q][h][w], w stride 1, h stride 128, q stride 16384,
// g stride 65536, b stride 4194304.  B=8, G=64, Q=4, H=W=128, Ho=Wo=64, L=4096.

// ---------------- Pass A: S[bg*4+kk] = sum_l mag(bg,kk,l) ----------------
__global__ __launch_bounds__(256) void qpool_sums(const float* __restrict__ x,
                                                  float* __restrict__ S) {
  const int bg = blockIdx.x;                       // 0..511 == b*64+g
  const float* xp = x + (size_t)bg * 65536;
  const int tid = threadIdx.x;
  float s0 = 0.f, s1 = 0.f, s2 = 0.f, s3 = 0.f;    // per-kk partial sums

#pragma unroll
  for (int r = 0; r < 8; ++r) {
    const int p  = r * 256 + tid;                  // window-pair index 0..2047
    const int h0 = (p >> 5) * 2;                   // 2*ho
    const int w0 = (p & 31) * 4;                   // 16B-aligned column
    float a00 = 0, a01 = 0, a02 = 0, a03 = 0;      // window0 kk0..3 sum(x^2)
    float a10 = 0, a11 = 0, a12 = 0, a13 = 0;      // window1 kk0..3
#pragma unroll
    for (int q = 0; q < 4; ++q) {
      const float* row = xp + q * 16384 + h0 * 128 + w0;
      const float4 v0 = *(const float4*)(row);         // row 2ho
      const float4 v1 = *(const float4*)(row + 128);   // row 2ho+1
      a00 += v0.x * v0.x; a01 += v0.y * v0.y;          // kk=(0,0),(0,1) win0
      a10 += v0.z * v0.z; a11 += v0.w * v0.w;          // win1
      a02 += v1.x * v1.x; a03 += v1.y * v1.y;          // kk=(1,0),(1,1) win0
      a12 += v1.z * v1.z; a13 += v1.w * v1.w;          // win1
    }
    s0 += __builtin_amdgcn_sqrtf(a00) + __builtin_amdgcn_sqrtf(a10);
    s1 += __builtin_amdgcn_sqrtf(a01) + __builtin_amdgcn_sqrtf(a11);
    s2 += __builtin_amdgcn_sqrtf(a02) + __builtin_amdgcn_sqrtf(a12);
    s3 += __builtin_amdgcn_sqrtf(a03) + __builtin_amdgcn_sqrtf(a13);
  }

  // wave32 reduction
  const int lane = tid & 31, wid = tid >> 5;
#pragma unroll
  for (int off = 16; off > 0; off >>= 1) {
    s0 += __shfl_down(s0, off, 32);
    s1 += __shfl_down(s1, off, 32);
    s2 += __shfl_down(s2, off, 32);
    s3 += __shfl_down(s3, off, 32);
  }
  __shared__ float red[8][4];
  if (lane == 0) { red[wid][0] = s0; red[wid][1] = s1; red[wid][2] = s2; red[wid][3] = s3; }
  __syncthreads();
  if (tid < 4) {
    float acc = 0.f;
#pragma unroll
    for (int w = 0; w < 8; ++w) acc += red[w][tid];
    S[bg * 4 + tid] = acc;
  }
}

// ------- Pass B: entropy via v_wmma_f32_16x16x4_f32, argmax_g, gather -------
// One block (4 waves) handles 16 output positions; wave w scans g in
// [16w, 16w+16). Per g, lane L contributes A[m][k] terms for item m = L&15
// (lanes 0-15: kk{0,1} from window row 0; lanes 16-31: kk{2,3} from row 1);
// WMMA with B=ones sums the 4 kk entropy terms (k-order irrelevant). D row m
// is column-broadcast: rows 0-7 in lanes 0-15, rows 8-15 in lanes 16-31.
// Cross-wave argmax combine in LDS preserves first-max semantics.
__global__ __launch_bounds__(128) void qpool_select(const float* __restrict__ x,
                                                    const float* __restrict__ S,
                                                    float* __restrict__ out) {
  const int tid  = threadIdx.x;
  const int wid  = tid >> 5;                       // 0..3 : g-range
  const int lane = tid & 31;
  const int b     = blockIdx.x >> 8;               // 0..7 (256 blocks per b)
  const int lbase = (blockIdx.x & 255) * 16;

  __shared__ float sS[256];                        // S for this b: 64 g x 4 kk
  __shared__ float sE[4][16];
  __shared__ int   sG[4][16];
  for (int i = tid; i < 256; i += 128) sS[i] = S[b * 256 + i];
  __syncthreads();

  const int m    = lane & 15;
  const int half = lane >> 4;                      // window row / kk pair
  const int l    = lbase + m;
  const int ho   = l >> 6, wo = l & 63;
  const float* xb = x + (size_t)b * 4194304 + (size_t)(2 * ho + half) * 128 + 2 * wo
                      + (size_t)(wid * 16) * 65536;

  const v2f bones = {1.f, 1.f};
  const int k8 = lane & 7;
  float bestE = -3.4e38f;
  int   bestG = wid * 16;

  for (int i = 0; i < 16; ++i) {
    const float* xg = xb + (size_t)i * 65536;
    float q0 = 0.f, q1 = 0.f;
#pragma unroll
    for (int q = 0; q < 4; ++q) {
      const float2 v = *(const float2*)(xg + q * 16384);
      q0 += v.x * v.x;
      q1 += v.y * v.y;
    }
    __builtin_prefetch(xg + 65536, 0, 0);          // next g plane -> global_prefetch_b8

    const int g = wid * 16 + i;
    const float mag0 = __builtin_amdgcn_sqrtf(q0);
    const float mag1 = __builtin_amdgcn_sqrtf(q1);
    const float inv0 = __builtin_amdgcn_rcpf(sS[g * 4 + half * 2 + 0] + QP_EPS);
    const float inv1 = __builtin_amdgcn_rcpf(sS[g * 4 + half * 2 + 1] + QP_EPS);
    const float p0 = mag0 * inv0, p1 = mag1 * inv1;

    v2f a;                                         // -p*ln(p+eps), ln = log2*ln2
    a.x = -p0 * (LN2F * __builtin_amdgcn_logf(p0 + QP_EPS));
    a.y = -p1 * (LN2F * __builtin_amdgcn_logf(p1 + QP_EPS));

    v8f c = {};
    // D[m][n] = sum_k A[m][k] * 1 : the 4-term entropy sum per item.
    v8f d = __builtin_amdgcn_wmma_f32_16x16x4_f32(
        /*neg_a=*/false, a, /*neg_b=*/false, bones,
        /*c_mod=*/(short)0, c, /*reuse_a=*/false, /*reuse_b=*/false);

    float e = d[0];
    e = (k8 == 1) ? d[1] : e;
    e = (k8 == 2) ? d[2] : e;
    e = (k8 == 3) ? d[3] : e;
    e = (k8 == 4) ? d[4] : e;
    e = (k8 == 5) ? d[5] : e;
    e = (k8 == 6) ? d[6] : e;
    e = (k8 == 7) ? d[7] : e;

    const bool upd = e > bestE;                    // strict > == first-max
    bestE = upd ? e : bestE;
    bestG = upd ? g : bestG;
  }

  // Owner lanes {0-7, 16-23} hold items {k8, k8+8}.
  const bool owner = ((lane & 8) == 0);
  const int  item  = k8 + half * 8;
  if (owner) { sE[wid][item] = bestE; sG[wid][item] = bestG; }
  __syncthreads();

  if (wid == 0 && owner) {
    float e  = sE[0][item];
    int   gb = sG[0][item];
#pragma unroll
    for (int w = 1; w < 4; ++w) {                  // ascending g-range: ties keep lower g
      const bool u = sE[w][item] > e;
      e  = u ? sE[w][item] : e;
      gb = u ? sG[w][item] : gb;
    }

    const int li  = lbase + item;
    const int iho = li >> 6, iwo = li & 63;
    const float* xs = x + (size_t)(b * 64 + gb) * 65536
                        + (size_t)(2 * iho) * 128 + 2 * iwo;
    float o[16];
#pragma unroll
    for (int q = 0; q < 4; ++q) {
      const float2 r0 = *(const float2*)(xs + q * 16384);         // row 2ho
      const float2 r1 = *(const float2*)(xs + q * 16384 + 128);   // row 2ho+1
      o[q * 4 + 0] = r0.x;   // kk=(0,0)
      o[q * 4 + 1] = r0.y;   // kk=(0,1)
      o[q * 4 + 2] = r1.x;   // kk=(1,0)
      o[q * 4 + 3] = r1.y;   // kk=(1,1)
    }
    float* op = out + (size_t)b * 65536 + (size_t)li * 16;  // flat: l*16+q*4+kk
#pragma unroll
    for (int t = 0; t < 4; ++t)
      *(float4*)(op + 4 * t) = make_float4(o[4 * t], o[4 * t + 1], o[4 * t + 2], o[4 * t + 3]);
  }
}

extern "C" void kernel_launch(void* const* d_in, const int* in_sizes, int n_in,
                              void* d_out, int out_size, void* d_ws, size_t ws_size,
                              hipStream_t stream) {
  (void)in_sizes; (void)n_in; (void)out_size; (void)ws_size;
  const float* x = (const float*)d_in[0];
  float* S   = (float*)d_ws;     // 2048 floats (8 KB)
  float* out = (float*)d_out;    // 524288 floats

  qpool_sums<<<dim3(512), dim3(256), 0, stream>>>(x, S);
  qpool_select<<<dim3(2048), dim3(128), 0, stream>>>(x, S, out);
}